// AnomalyTransformer_86294482911876
// MI455X (gfx1250) — compile-verified
//
#include <hip/hip_runtime.h>
#include <math.h>

// Problem constants (from reference)
#define BATCH 256
#define SEQL  100
#define CIN   55
#define DMODEL 512
#define NHEAD 8
#define DK    64
#define NLAYER 3
#define DFFN  64
#define LPAD  112            // SEQL padded to 7 x 16
#define MROWS (BATCH * SEQL) // 25600, multiple of 16

typedef float v2f __attribute__((ext_vector_type(2)));
typedef float v8f __attribute__((ext_vector_type(8)));
typedef int   v4i __attribute__((ext_vector_type(4)));

#define GLOBAL_AS __attribute__((address_space(1)))
#define LDS_AS __attribute__((address_space(3)))

// Async global->LDS staging (CDNA5 ASYNCcnt path). Guarded: falls back to
// load/store staging if the builtin is unavailable on this toolchain.
#if __has_builtin(__builtin_amdgcn_global_load_async_to_lds_b128)
#define HAVE_ASYNC_LDS 1
__device__ __forceinline__ void async_copy16(const float* g, float* l) {
  __builtin_amdgcn_global_load_async_to_lds_b128(
      (GLOBAL_AS v4i*)const_cast<float*>(g), (LDS_AS v4i*)l, 0, 0);
}
__device__ __forceinline__ void async_wait0() {
#if __has_builtin(__builtin_amdgcn_s_wait_asynccnt)
  __builtin_amdgcn_s_wait_asynccnt(0);
#else
  asm volatile("s_wait_asynccnt 0x0" ::: "memory");
#endif
}
#else
#define HAVE_ASYNC_LDS 0
__device__ __forceinline__ void async_copy16(const float* g, float* l) {
  *(float4*)l = *(const float4*)g;
}
__device__ __forceinline__ void async_wait0() {}
#endif

// CDNA5 fp32 WMMA: D(16x16) = A(16x4) x B(4x16) + C, exact fp32.
__device__ __forceinline__ v8f wmma_f32(v2f a, v2f b, v8f c) {
  return __builtin_amdgcn_wmma_f32_16x16x4_f32(false, a, false, b, (short)0, c,
                                               false, false);
}

// ---------------------------------------------------------------------------
// Token embedding: circular conv1d(k=3, no bias) + positional embedding.
// ---------------------------------------------------------------------------
__global__ void embed_kernel(const float* __restrict__ x,
                             const float* __restrict__ tokw,
                             const float* __restrict__ pe,
                             float* __restrict__ h) {
  int idx = blockIdx.x * blockDim.x + threadIdx.x;
  if (idx >= MROWS * DMODEL) return;
  int d = idx % DMODEL;
  int l = (idx / DMODEL) % SEQL;
  int b = idx / (DMODEL * SEQL);
  int lm = (l + SEQL - 1) % SEQL;
  int lp = (l + 1) % SEQL;
  const float* x0 = x + ((size_t)b * SEQL + lm) * CIN;
  const float* x1 = x + ((size_t)b * SEQL + l) * CIN;
  const float* x2 = x + ((size_t)b * SEQL + lp) * CIN;
  const float* w = tokw + (size_t)d * CIN * 3;
  float acc = 0.f;
#pragma unroll 5
  for (int c = 0; c < CIN; ++c) {
    acc = fmaf(x0[c], w[c * 3 + 0], acc);
    acc = fmaf(x1[c], w[c * 3 + 1], acc);
    acc = fmaf(x2[c], w[c * 3 + 2], acc);
  }
  h[idx] = acc + pe[l * DMODEL + d];
}

// ---------------------------------------------------------------------------
// LDS-staged GEMM for wide N: out[M,N] = A[M,K] * W[N,K]^T + bias (+epilogue)
// Requirements: M%16==0, N%256==0, K%32==0.
// Block = 4 waves: one 16-row M tile x 256 cols (wave w owns cols [w*64,+64)).
// Per 32-deep K chunk: stage A(16x32) and W(256x32) into LDS with async
// global->LDS b128 copies (ASYNCcnt), then feed WMMA fragments from LDS.
// Row stride padded to 36 floats to avoid 64-bank conflicts on ds_load_b64.
//   EPI 0: out = acc + bias
//   EPI 1: out = acc + bias + res
//   EPI 2: out = gelu_exact(acc + bias)
// ---------------------------------------------------------------------------
#define KC 32
#define LDSTRIDE 36
template <int EPI>
__global__ void gemm_lds_kernel(const float* __restrict__ A,
                                const float* __restrict__ W,
                                const float* __restrict__ bias,
                                const float* __restrict__ res,
                                float* __restrict__ out, int M, int N, int K) {
  __shared__ float As[16 * LDSTRIDE];
  __shared__ float Ws[256 * LDSTRIDE];

  int nb = N >> 8;
  int bm = blockIdx.x / nb;
  int bn = blockIdx.x % nb;
  int row0 = bm << 4;
  int colBase = bn << 8;
  int w = threadIdx.y, lane = threadIdx.x;
  int tid = w * 32 + lane;
  int hi = lane >> 4, lo = lane & 15;
  int col0 = colBase + w * 64;

  v8f acc[4] = {};
#pragma unroll 1
  for (int kc = 0; kc < K; kc += KC) {
    // Stage A tile (16 x 32): 128 x b128 async copies, one per thread
    {
      int r = tid >> 3;
      int c = (tid & 7) << 2;
      async_copy16(A + (size_t)(row0 + r) * K + kc + c, &As[r * LDSTRIDE + c]);
    }
    // Stage W tile (256 x 32): 2048 x b128 async copies, 16 per thread
#pragma unroll 1
    for (int i = tid; i < 256 * 8; i += 128) {
      int r = i >> 3;
      int c = (i & 7) << 2;
      async_copy16(W + (size_t)(colBase + r) * K + kc + c,
                   &Ws[r * LDSTRIDE + c]);
    }
    async_wait0();
    __syncthreads();
#pragma unroll 4
    for (int k0 = 0; k0 < KC; k0 += 4) {
      const float* ap = &As[lo * LDSTRIDE + k0 + 2 * hi];
      v2f a = {ap[0], ap[1]};
#pragma unroll
      for (int j = 0; j < 4; ++j) {
        const float* wp = &Ws[(w * 64 + j * 16 + lo) * LDSTRIDE + k0 + 2 * hi];
        v2f bf = {wp[0], wp[1]};
        acc[j] = wmma_f32(a, bf, acc[j]);
      }
    }
    __syncthreads();
  }

#pragma unroll
  for (int j = 0; j < 4; ++j) {
    int n = col0 + j * 16 + lo;
    float bv = bias[n];
#pragma unroll
    for (int r = 0; r < 8; ++r) {
      int m = row0 + r + 8 * hi;
      size_t oi = (size_t)m * N + n;
      float v = acc[j][r] + bv;
      if (EPI == 1) v += res[oi];
      if (EPI == 2) v = 0.5f * v * (1.0f + erff(v * 0.70710678118654752f));
      out[oi] = v;
    }
  }
}

// ---------------------------------------------------------------------------
// Direct GEMM for small N (<= 64): one wave per 16 x min(N,64) tile.
// N may be ragged (masked). K % 4 == 0. Unroll bounded to avoid spills.
// ---------------------------------------------------------------------------
template <int EPI>
__global__ void gemm_small_kernel(const float* __restrict__ A,
                                  const float* __restrict__ W,
                                  const float* __restrict__ bias,
                                  const float* __restrict__ res,
                                  float* __restrict__ out, int M, int N,
                                  int K) {
  int totalTiles = M >> 4;
  int t = blockIdx.x * 4 + threadIdx.y;  // wave-uniform
  if (t >= totalTiles) return;
  int row0 = t << 4;
  int lane = threadIdx.x;
  int hi = lane >> 4, lo = lane & 15;

  v8f acc[4] = {};
  const float* arow = A + (size_t)(row0 + lo) * K + 2 * hi;
#pragma unroll 2
  for (int k0 = 0; k0 < K; k0 += 4) {
    v2f a = {arow[k0], arow[k0 + 1]};
#pragma unroll
    for (int j = 0; j < 4; ++j) {
      int n = j * 16 + lo;
      v2f bf;
      if (n < N) {
        const float* wp = W + (size_t)n * K + k0 + 2 * hi;
        bf = v2f{wp[0], wp[1]};
      } else {
        bf = v2f{0.f, 0.f};
      }
      acc[j] = wmma_f32(a, bf, acc[j]);
    }
  }

#pragma unroll
  for (int j = 0; j < 4; ++j) {
    int n = j * 16 + lo;
    if (n >= N) continue;
    float bv = bias[n];
#pragma unroll
    for (int r = 0; r < 8; ++r) {
      int m = row0 + r + 8 * hi;
      size_t oi = (size_t)m * N + n;
      float v = acc[j][r] + bv;
      if (EPI == 1) v += res[oi];
      if (EPI == 2) v = 0.5f * v * (1.0f + erff(v * 0.70710678118654752f));
      out[oi] = v;
    }
  }
}

// ---------------------------------------------------------------------------
// Attention: per (b,h) block. scores = (Q Kh^T) * scale ; softmax ; P Vh.
// K/V staged into LDS via async global->LDS copies; pad rows zero-filled.
// ---------------------------------------------------------------------------
__global__ void attn_kernel(const float* __restrict__ Q,
                            const float* __restrict__ Kg,
                            const float* __restrict__ Vg,
                            float* __restrict__ O) {
  __shared__ float kbuf[LPAD * DK];
  __shared__ float vbuf[LPAD * DK];
  __shared__ float pbuf[4 * 16 * LPAD];

  const int b = blockIdx.x / NHEAD;
  const int h = blockIdx.x % NHEAD;
  const int lane = threadIdx.x;
  const int w = threadIdx.y;
  const int tid = w * 32 + lane;
  const int hi = lane >> 4, lo = lane & 15;
  const float scale = 0.125f;  // 1/sqrt(dk)

  // Zero-fill pad rows [SEQL, LPAD)
  for (int i = tid; i < (LPAD - SEQL) * DK; i += 128) {
    kbuf[SEQL * DK + i] = 0.f;
    vbuf[SEQL * DK + i] = 0.f;
  }
  // Async-copy the 100 valid rows (16 float4 per row)
  for (int i = tid; i < SEQL * (DK / 4); i += 128) {
    int s = i >> 4;
    int c = (i & 15) << 2;
    size_t g = ((size_t)b * SEQL + s) * DMODEL + h * DK + c;
    async_copy16(Kg + g, &kbuf[s * DK + c]);
    async_copy16(Vg + g, &vbuf[s * DK + c]);
  }
  async_wait0();
  __syncthreads();

  float* pb = pbuf + w * 16 * LPAD;

  for (int qt = w; qt < 7; qt += 4) {
    int l = qt * 16 + lo;
    if (l > SEQL - 1) l = SEQL - 1;  // clamp padded rows (discarded on store)
    const float* qrow = Q + ((size_t)b * SEQL + l) * DMODEL + h * DK + 2 * hi;
#pragma unroll 1
    for (int st = 0; st < 7; ++st) {
      v8f acc = {};
      const float* krow = kbuf + (st * 16 + lo) * DK + 2 * hi;
#pragma unroll 4
      for (int e0 = 0; e0 < DK; e0 += 4) {
        v2f a = {qrow[e0], qrow[e0 + 1]};
        v2f bf = {krow[e0], krow[e0 + 1]};
        acc = wmma_f32(a, bf, acc);
      }
#pragma unroll
      for (int r = 0; r < 8; ++r)
        pb[(r + 8 * hi) * LPAD + st * 16 + lo] = acc[r];
    }

    if (lane < 16) {
      float* row = pb + lane * LPAD;
      float mx = -1e30f;
      for (int s = 0; s < SEQL; ++s) {
        float v = row[s] * scale;
        if (v > mx) mx = v;
      }
      float sum = 0.f;
      for (int s = 0; s < LPAD; ++s) {
        float p = (s < SEQL) ? expf(row[s] * scale - mx) : 0.f;
        row[s] = p;
        sum += p;
      }
      float inv = 1.f / sum;
      for (int s = 0; s < LPAD; ++s) row[s] *= inv;
    }

#pragma unroll 1
    for (int te = 0; te < 4; ++te) {
      v8f acc = {};
#pragma unroll 4
      for (int s0 = 0; s0 < LPAD; s0 += 4) {
        const float* pp = pb + lo * LPAD + s0 + 2 * hi;
        v2f a = {pp[0], pp[1]};
        const float* vp = vbuf + (s0 + 2 * hi) * DK + te * 16 + lo;
        v2f bf = {vp[0], vp[DK]};
        acc = wmma_f32(a, bf, acc);
      }
#pragma unroll
      for (int r = 0; r < 8; ++r) {
        int ls = qt * 16 + r + 8 * hi;
        if (ls < SEQL)
          O[((size_t)b * SEQL + ls) * DMODEL + h * DK + te * 16 + lo] = acc[r];
      }
    }
  }
}

// ---------------------------------------------------------------------------
// Orchestration
// ---------------------------------------------------------------------------
extern "C" void kernel_launch(void* const* d_in, const int* in_sizes, int n_in,
                              void* d_out, int out_size, void* d_ws,
                              size_t ws_size, hipStream_t stream) {
  const float* x      = (const float*)d_in[0];
  const float* tok_w  = (const float*)d_in[1];
  const float* pe     = (const float*)d_in[2];
  const float* Wq     = (const float*)d_in[3];
  const float* bq     = (const float*)d_in[4];
  const float* Wk     = (const float*)d_in[5];
  const float* bk     = (const float*)d_in[6];
  const float* Wv     = (const float*)d_in[7];
  const float* bv     = (const float*)d_in[8];
  // d_in[9]/d_in[10]: Wsig/bsig feed only the unused prior branch: skipped
  const float* Wo     = (const float*)d_in[11];
  const float* bo     = (const float*)d_in[12];
  const float* W1     = (const float*)d_in[13];
  const float* b1     = (const float*)d_in[14];
  const float* W2     = (const float*)d_in[15];
  const float* b2     = (const float*)d_in[16];
  const float* proj_w = (const float*)d_in[17];
  const float* proj_b = (const float*)d_in[18];
  float* out = (float*)d_out;

  const size_t SZ = (size_t)MROWS * DMODEL;
  float* ws = (float*)d_ws;
  float* hA = ws;
  float* hB = hA + SZ;
  float* q  = hB + SZ;  // Q, then attention output (in-place)
  float* kb = q + SZ;
  float* vb = kb + SZ;
  float* y  = vb + SZ;  // FFN intermediate [MROWS, DFFN]

  dim3 gblk(32, 4);
  const int gridBig = (MROWS >> 4) * (DMODEL >> 8);   // 3200 blocks
  const int gridSmall = ((MROWS >> 4) + 3) / 4;       // 400 blocks

  {
    int total = MROWS * DMODEL;
    embed_kernel<<<(total + 255) / 256, 256, 0, stream>>>(x, tok_w, pe, hA);
  }

  for (int l = 0; l < NLAYER; ++l) {
    const size_t wOff = (size_t)l * DMODEL * DMODEL;
    const size_t bOff = (size_t)l * DMODEL;
    gemm_lds_kernel<0><<<gridBig, gblk, 0, stream>>>(
        hA, Wq + wOff, bq + bOff, nullptr, q, MROWS, DMODEL, DMODEL);
    gemm_lds_kernel<0><<<gridBig, gblk, 0, stream>>>(
        hA, Wk + wOff, bk + bOff, nullptr, kb, MROWS, DMODEL, DMODEL);
    gemm_lds_kernel<0><<<gridBig, gblk, 0, stream>>>(
        hA, Wv + wOff, bv + bOff, nullptr, vb, MROWS, DMODEL, DMODEL);
    attn_kernel<<<BATCH * NHEAD, gblk, 0, stream>>>(q, kb, vb, q);
    gemm_lds_kernel<1><<<gridBig, gblk, 0, stream>>>(
        q, Wo + wOff, bo + bOff, hA, hB, MROWS, DMODEL, DMODEL);
    gemm_small_kernel<2><<<gridSmall, gblk, 0, stream>>>(
        hB, W1 + (size_t)l * DFFN * DMODEL, b1 + (size_t)l * DFFN, nullptr, y,
        MROWS, DFFN, DMODEL);
    gemm_lds_kernel<1><<<gridBig, gblk, 0, stream>>>(
        y, W2 + (size_t)l * DMODEL * DFFN, b2 + bOff, hB, hA, MROWS, DMODEL,
        DFFN);
  }

  gemm_small_kernel<0><<<gridSmall, gblk, 0, stream>>>(
      hA, proj_w, proj_b, nullptr, out, MROWS, CIN, DMODEL);
}